// Embedding_43774306681498
// MI455X (gfx1250) — compile-verified
//
#include <hip/hip_runtime.h>

typedef __attribute__((ext_vector_type(2))) float v2f;
typedef __attribute__((ext_vector_type(8))) float v8f;

#define ZD_ 64
#define PD_ 16
#define DH_ 128

// ---------------- Kernel A: hid = relu([z | pde] @ W1^T + b1), [64 x 128] ----
__global__ void hypernet_hidden_kernel(const float* __restrict__ z_all,
                                       const float* __restrict__ W1,
                                       const float* __restrict__ b1,
                                       const float* __restrict__ pde,
                                       float* __restrict__ hid) {
    int tid = blockIdx.x * blockDim.x + threadIdx.x; // 0..8191
    int r = tid >> 7;       // hk row 0..63
    int c = tid & 127;      // hidden unit 0..127
    const float* wrow = W1 + c * (ZD_ + PD_);
    const float* zrow = z_all + r * ZD_;
    float acc = b1[c];
#pragma unroll 8
    for (int d = 0; d < ZD_; ++d) acc += zrow[d] * wrow[d];
#pragma unroll
    for (int p = 0; p < PD_; ++p) acc += pde[p] * wrow[ZD_ + p];
    hid[tid] = acc > 0.0f ? acc : 0.0f;
}

// ---------------- Kernel B: fused (hid @ W2^T + b2) x outer(u) ---------------
// Grid: (256, 3) blocks of 128 threads (4 waves).
//   blockIdx.x = oi-group g (16 consecutive (ob,ib) pairs)
//   blockIdx.y = kq, handling kernel positions k = 3*kq .. 3*kq+2
//   wave m     = M-tile of 16 hk rows [16m, 16m+16)
// Each wave keeps 3 v8f accumulators and runs 3 x 32 V_WMMA_F32_16X16X4_F32
// (K=128 in chunks of 4), then the fused outer-product epilogue writes
// 27 contiguous floats per (o,i) slot.
__global__ void __launch_bounds__(128) hyper_w2_outer_kernel(
    const float* __restrict__ hid,   // [64,128]
    const float* __restrict__ W2,    // [36864,128]  row j = oi*9 + k
    const float* __restrict__ b2,    // [36864]
    const float* __restrict__ U,     // [512,512,9]
    float* __restrict__ OUT)         // [512,512,9,9]
{
    const int g    = blockIdx.x;        // oi-group 0..255
    const int kq   = blockIdx.y;        // k-triplet 0..2
    const int k0   = kq * 3;
    const int wv   = threadIdx.x >> 5;  // wave id = M-tile 0..3
    const int lane = threadIdx.x & 31;
    const int ln   = lane & 15;         // N / M-within-half index
    const int half = lane >> 4;         // K-pair / M+8 selector
    const int m0   = wv * 16;

    // A-frag: lane(ln,half) holds hid[m0+ln][kc + 2*half + {0,1}]  (ISA 16x4 layout)
    const float* Abase = hid + (size_t)(m0 + ln) * DH_ + 2 * half;
    // B-frag: lane(ln,half) holds W2[(g*16+ln)*9 + k0+kt][kc + 2*half + {0,1}]
    const float* Bbase = W2 + (size_t)((g * 16 + ln) * 9 + k0) * DH_ + 2 * half;

    v8f acc[3] = {};
    for (int kc = 0; kc < DH_; kc += 4) {
        v2f a = *(const v2f*)(Abase + kc);       // reused across the 3 k's
#pragma unroll
        for (int kt = 0; kt < 3; ++kt) {
            v2f b = *(const v2f*)(Bbase + kt * DH_ + kc);
            acc[kt] = __builtin_amdgcn_wmma_f32_16x16x4_f32(
                false, a, false, b, (short)0, acc[kt], false, false);
        }
    }

    const int oi = g * 16 + ln;
    float bb[3];
#pragma unroll
    for (int kt = 0; kt < 3; ++kt) bb[kt] = b2[oi * 9 + k0 + kt];

    // D layout: VGPR r -> M = r (lanes 0-15) / r+8 (lanes 16-31), N = lane&15.
#pragma unroll
    for (int r = 0; r < 8; ++r) {
        int hk = m0 + r + half * 8;                 // hypernet row = h*8 + kb
        int o  = (hk >> 3) * 64 + (oi >> 6);        // h*64 + ob
        int i  = (hk & 7) * 64 + (oi & 63);         // kb*64 + ib
        size_t base = (size_t)o * 512 + (size_t)i;
        const float* up = U + base * 9;
        float uv[9];
#pragma unroll
        for (int l = 0; l < 9; ++l) uv[l] = up[l];  // loaded once, reused for 3 k
        float* op = OUT + base * 81 + (size_t)k0 * 9;
#pragma unroll
        for (int kt = 0; kt < 3; ++kt) {
            float wval = acc[kt][r] + bb[kt];
#pragma unroll
            for (int l = 0; l < 9; ++l) op[kt * 9 + l] = wval * uv[l];
        }
    }
}

extern "C" void kernel_launch(void* const* d_in, const int* in_sizes, int n_in,
                              void* d_out, int out_size, void* d_ws, size_t ws_size,
                              hipStream_t stream) {
    (void)in_sizes; (void)n_in; (void)out_size; (void)ws_size;
    const float* z_all = (const float*)d_in[0];
    const float* W1    = (const float*)d_in[1];
    const float* b1    = (const float*)d_in[2];
    const float* W2    = (const float*)d_in[3];
    const float* b2    = (const float*)d_in[4];
    const float* U     = (const float*)d_in[5];
    const float* pde   = (const float*)d_in[6];
    float* out = (float*)d_out;
    float* hid = (float*)d_ws;   // 64*128*4 = 32 KB scratch

    hypernet_hidden_kernel<<<32, 256, 0, stream>>>(z_all, W1, b1, pde, hid);
    hyper_w2_outer_kernel<<<dim3(256, 3), 128, 0, stream>>>(hid, W2, b2, U, out);
}